// QwenForCausalLM_64390149701651
// MI455X (gfx1250) — compile-verified
//
#include <hip/hip_runtime.h>
#include <hip/hip_bf16.h>
#include <math.h>

typedef float v2f __attribute__((ext_vector_type(2)));
typedef float v8f __attribute__((ext_vector_type(8)));

#define NLAYER 4
#define BB 8
#define SS 4096
#define HH 2048
#define NHEAD 16
#define KVH 2
#define DD 128
#define GG 8
#define IW 5632
#define VOCAB 32000
#define QKV_N 2560
#define SP 2064  /* 129*16 padded sequence */

// Pair-interleaved padded A layout: element (k, row) of a [K][16] matrix lives
// at (k>>1)*32 + row*2 + (k&1). Rows 8..15 are kept zero by the producers.
__device__ __forceinline__ int aidx(int k, int r) {
  return ((k >> 1) << 5) + (r << 1) + (k & 1);
}

__device__ __forceinline__ v8f wmma_f32_4(v2f a, v2f b, v8f c) {
  return __builtin_amdgcn_wmma_f32_16x16x4_f32(false, a, false, b, (short)0, c,
                                               false, false);
}

// ---------------------------------------------------------------- embed gather
__global__ __launch_bounds__(256) void embed_kernel(
    const int* __restrict__ tok, const float* __restrict__ emb,
    float* __restrict__ h) {
  const int b = blockIdx.x;
  const int t = tok[b];  // (B, T=1)
  const float* e = emb + (size_t)t * HH;
  const float sc = 45.254833995939045f;  // sqrt(2048)
  for (int k = threadIdx.x; k < HH; k += 256) h[(size_t)b * HH + k] = e[k] * sc;
}

// ------------------------------------------- rmsnorm -> padded/interleaved A
__global__ __launch_bounds__(256) void rmsnorm_kernel(
    const float* __restrict__ h, const float* __restrict__ w,
    float* __restrict__ xt) {
  const int b = blockIdx.x;
  __shared__ float red[256];
  const float* hb = h + (size_t)b * HH;
  float s = 0.0f;
  for (int k = threadIdx.x; k < HH; k += 256) { float v = hb[k]; s += v * v; }
  red[threadIdx.x] = s;
  __syncthreads();
  for (int st = 128; st > 0; st >>= 1) {
    if (threadIdx.x < st) red[threadIdx.x] += red[threadIdx.x + st];
    __syncthreads();
  }
  const float r = rsqrtf(red[0] / (float)HH + 1e-6f);
  for (int k = threadIdx.x; k < HH; k += 256) {
    const int i = aidx(k, b);
    xt[i] = hb[k] * r * w[k];
    xt[i + 16] = 0.0f;  // pad row b+8
  }
}

// ------------------------------------------------- WMMA GEMM: [16xK] x [KxN]
// At: padded/interleaved activations. EPI: 0 store, 1 silu-store, 2 residual+=.
// WT=false: W row-major [K][N]. WT=true: W is [N][K] (embed for logits).
template <int K, int EPI, bool WT>
__global__ __launch_bounds__(256) void gemm_kernel(
    const float* __restrict__ At, const float* __restrict__ W,
    float* __restrict__ out, int N) {
  const int wave = threadIdx.x >> 5;
  const int lane = threadIdx.x & 31;
  const int tile = blockIdx.x * 8 + wave;
  if (tile * 16 >= N) return;  // wave-uniform
  const int n0 = tile * 16;
  const int l16 = lane & 15;
  const int kc = (lane >> 4) << 1;  // 0 or 2
  const float* __restrict__ ap = At + kc * 16 + l16 * 2;
  const float* __restrict__ wp =
      WT ? (W + (size_t)(n0 + l16) * K + kc) : (W + (size_t)kc * N + n0 + l16);
  v8f acc = {};
  v2f a = *(const v2f*)ap;
  v2f b;
  if (WT) {
    b.x = __builtin_nontemporal_load(wp);
    b.y = __builtin_nontemporal_load(wp + 1);
  } else {
    b.x = __builtin_nontemporal_load(wp);
    b.y = __builtin_nontemporal_load(wp + N);
  }
#pragma unroll 4
  for (int k = 4; k < K; k += 4) {
    ap += 64;
    const v2f an = *(const v2f*)ap;
    v2f bn;
    if (WT) {
      wp += 4;
      bn.x = __builtin_nontemporal_load(wp);
      bn.y = __builtin_nontemporal_load(wp + 1);
    } else {
      wp += (size_t)4 * N;
      bn.x = __builtin_nontemporal_load(wp);
      bn.y = __builtin_nontemporal_load(wp + N);
    }
    acc = wmma_f32_4(a, b, acc);
    a = an;
    b = bn;
  }
  acc = wmma_f32_4(a, b, acc);
  if ((lane >> 4) == 0) {
    const int n = n0 + l16;
#pragma unroll
    for (int j = 0; j < 8; ++j) {
      float v = acc[j];
      if (EPI == 1) v = v / (1.0f + expf(-v));  // silu
      float* o = out + (size_t)j * N + n;
      if (EPI == 2) *o += v; else *o = v;
    }
  }
}

// ------------------------------------------------------------------ RoPE split
__global__ __launch_bounds__(256) void rope_kernel(
    const float* __restrict__ qkv, const float* __restrict__ rc,
    const float* __restrict__ rs, const int* __restrict__ pos_p,
    float* __restrict__ qt, float* __restrict__ kn, float* __restrict__ vn) {
  const int b = blockIdx.x;
  const int pos = pos_p[0];
  const float* c = rc + (size_t)pos * (DD / 2);
  const float* s = rs + (size_t)pos * (DD / 2);
  const float* q = qkv + (size_t)b * QKV_N;
  const float scale = 0.08838834764831845f;  // 128^-0.5
  for (int idx = threadIdx.x; idx < NHEAD * (DD / 2); idx += 256) {
    const int hh = idx >> 6, i = idx & 63;
    const float x1 = q[hh * DD + i], x2 = q[hh * DD + 64 + i];
    const float o1 = (x1 * c[i] - x2 * s[i]) * scale;
    const float o2 = (x1 * s[i] + x2 * c[i]) * scale;
    const int kv = hh >> 3, g = hh & 7;  // head = kv*G + g
    float* qb = qt + (size_t)(b * KVH + kv) * DD * 16;
    const int i0 = aidx(i, g), i1 = aidx(i + 64, g);
    qb[i0] = o1;
    qb[i0 + 16] = 0.0f;
    qb[i1] = o2;
    qb[i1 + 16] = 0.0f;
  }
  for (int idx = threadIdx.x; idx < KVH * (DD / 2); idx += 256) {
    const int kv = idx >> 6, i = idx & 63;
    const float* kk = q + NHEAD * DD + kv * DD;
    const float x1 = kk[i], x2 = kk[64 + i];
    kn[(size_t)(b * KVH + kv) * DD + i] = x1 * c[i] - x2 * s[i];
    kn[(size_t)(b * KVH + kv) * DD + 64 + i] = x1 * s[i] + x2 * c[i];
  }
  for (int idx = threadIdx.x; idx < KVH * DD; idx += 256)
    vn[(size_t)b * KVH * DD + idx] = q[(NHEAD + KVH) * DD + idx];
}

// ------------------------------------- scores: Q[16x128] x K^T -> pt[s][16]
__global__ __launch_bounds__(256) void scores_kernel(
    const float* __restrict__ qt, const float* __restrict__ kcache,
    const float* __restrict__ kn, float* __restrict__ pt,
    const int* __restrict__ pos_p) {
  const int gw = blockIdx.x * 8 + (threadIdx.x >> 5);
  if (gw >= BB * KVH * (SP / 16)) return;
  const int tile = gw % (SP / 16);
  const int bk = gw / (SP / 16);
  const int pos = pos_p[0];
  const int lane = threadIdx.x & 31;
  const int l16 = lane & 15;
  const int kc = (lane >> 4) << 1;
  const int s = tile * 16 + l16;
  const float* __restrict__ ap = qt + (size_t)bk * DD * 16 + kc * 16 + l16 * 2;
  // k_caches[b][s][kv][d]; s==pos -> new key; s>pos -> dummy (masked below)
  const float* __restrict__ krow =
      (s < pos) ? (kcache + (size_t)(bk >> 1) * SS * (KVH * DD) +
                   (size_t)s * (KVH * DD) + (size_t)(bk & 1) * DD)
                : (kn + (size_t)bk * DD);
  v8f acc = {};
#pragma unroll 4
  for (int k = 0; k < DD; k += 4) {
    const v2f a = *(const v2f*)(ap + k * 16);
    v2f b;
    b.x = __builtin_nontemporal_load(krow + k + kc);
    b.y = __builtin_nontemporal_load(krow + k + kc + 1);
    acc = wmma_f32_4(a, b, acc);
  }
  if ((lane >> 4) == 0) {
    float* dst = pt + (size_t)bk * SP * 16;
    const bool valid = (s <= pos);
#pragma unroll
    for (int j = 0; j < 8; ++j) {
      const int i = aidx(s, j);
      dst[i] = valid ? acc[j] : -1.0e9f;
      dst[i + 16] = 0.0f;  // pad rows g=8..15
    }
  }
}

// -------------------------------------------------- softmax over s per (b,kv,g)
__global__ __launch_bounds__(256) void softmax_kernel(float* __restrict__ pt) {
  const int bk = blockIdx.x;
  float* p = pt + (size_t)bk * SP * 16;
  __shared__ float red[GG][256];
  const int t = threadIdx.x;
  float mx[GG], sm[GG];
#pragma unroll
  for (int g = 0; g < GG; ++g) mx[g] = -3.0e38f;
  for (int s = t; s < SP; s += 256)
#pragma unroll
    for (int g = 0; g < GG; ++g) mx[g] = fmaxf(mx[g], p[aidx(s, g)]);
#pragma unroll
  for (int g = 0; g < GG; ++g) red[g][t] = mx[g];
  __syncthreads();
  for (int st = 128; st > 0; st >>= 1) {
    if (t < st)
#pragma unroll
      for (int g = 0; g < GG; ++g) red[g][t] = fmaxf(red[g][t], red[g][t + st]);
    __syncthreads();
  }
#pragma unroll
  for (int g = 0; g < GG; ++g) { mx[g] = red[g][0]; sm[g] = 0.0f; }
  __syncthreads();
  for (int s = t; s < SP; s += 256)
#pragma unroll
    for (int g = 0; g < GG; ++g) {
      const float e = expf(p[aidx(s, g)] - mx[g]);
      p[aidx(s, g)] = e;
      sm[g] += e;
    }
#pragma unroll
  for (int g = 0; g < GG; ++g) red[g][t] = sm[g];
  __syncthreads();
  for (int st = 128; st > 0; st >>= 1) {
    if (t < st)
#pragma unroll
      for (int g = 0; g < GG; ++g) red[g][t] += red[g][t + st];
    __syncthreads();
  }
#pragma unroll
  for (int g = 0; g < GG; ++g) sm[g] = 1.0f / red[g][0];
  for (int s = t; s < SP; s += 256)
#pragma unroll
    for (int g = 0; g < GG; ++g) p[aidx(s, g)] *= sm[g];
}

// --------------------------------------------- attn out: P[16xSP] x V[SPx128]
__global__ __launch_bounds__(256) void attn_kernel(
    const float* __restrict__ pt, const float* __restrict__ vcache,
    const float* __restrict__ vn, float* __restrict__ at,
    const int* __restrict__ pos_p) {
  const int gw = blockIdx.x * 8 + (threadIdx.x >> 5);
  if (gw >= BB * KVH * (DD / 16)) return;
  const int dt = gw & 7;
  const int bk = gw >> 3;
  const int pos = pos_p[0];
  const int lane = threadIdx.x & 31;
  const int l16 = lane & 15;
  const int kc = (lane >> 4) << 1;
  const int d0 = dt * 16;
  const float* __restrict__ ap = pt + (size_t)bk * SP * 16 + kc * 16 + l16 * 2;
  const float* vcb =
      vcache + (size_t)(bk >> 1) * SS * (KVH * DD) + (size_t)(bk & 1) * DD;
  const float* vnr = vn + (size_t)bk * DD;
  v8f acc = {};
#pragma unroll 4
  for (int k = 0; k < SP; k += 4) {
    const int s0 = k + kc;
    const v2f a = *(const v2f*)(ap + k * 16);
    // s==pos uses new value row; s>pos rows have p==0 so any finite row works
    const float* r0 = (s0 < pos) ? vcb + (size_t)s0 * (KVH * DD) : vnr;
    const float* r1 = (s0 + 1 < pos) ? vcb + (size_t)(s0 + 1) * (KVH * DD) : vnr;
    v2f b;
    b.x = __builtin_nontemporal_load(r0 + d0 + l16);
    b.y = __builtin_nontemporal_load(r1 + d0 + l16);
    acc = wmma_f32_4(a, b, acc);
  }
  if ((lane >> 4) == 0) {
    const int b = bk >> 1, kv = bk & 1;
    const int d = d0 + l16;
#pragma unroll
    for (int j = 0; j < 8; ++j) {
      const int row = (kv * GG + j) * DD + d;
      const int i = aidx(row, b);
      at[i] = acc[j];
      at[i + 16] = 0.0f;  // pad row b+8
    }
  }
}

// ------------------------------------------------------- t = silu(gate) * up
__global__ __launch_bounds__(256) void gateup_kernel(
    const float* __restrict__ g, const float* __restrict__ u,
    float* __restrict__ tt) {
  const int i = blockIdx.x * 256 + threadIdx.x;
  if (i < BB * IW) {
    const int b = i / IW, n = i % IW;
    const int o = aidx(n, b);
    tt[o] = g[i] * u[i];  // silu already applied in gate GEMM
    tt[o + 16] = 0.0f;    // pad row b+8
  }
}

// ------------------------------------------------------------ argmax per batch
__global__ __launch_bounds__(256) void argmax_kernel(
    const float* __restrict__ logits, int* __restrict__ out) {
  const int b = blockIdx.x;
  const float* lg = logits + (size_t)b * VOCAB;
  __shared__ float vmax[256];
  __shared__ int vidx[256];
  const int t = threadIdx.x;
  float m = -3.0e38f;
  int mi = 0;
  for (int v = t; v < VOCAB; v += 256) {
    const float x = lg[v];
    if (x > m) { m = x; mi = v; }
  }
  vmax[t] = m; vidx[t] = mi;
  __syncthreads();
  for (int st = 128; st > 0; st >>= 1) {
    if (t < st) {
      if (vmax[t + st] > vmax[t] ||
          (vmax[t + st] == vmax[t] && vidx[t + st] < vidx[t])) {
        vmax[t] = vmax[t + st];
        vidx[t] = vidx[t + st];
      }
    }
    __syncthreads();
  }
  if (t == 0) out[b] = vidx[0];
}

extern "C" void kernel_launch(void* const* d_in, const int* in_sizes, int n_in,
                              void* d_out, int out_size, void* d_ws,
                              size_t ws_size, hipStream_t stream) {
  const int* tok = (const int*)d_in[0];
  const int* pos = (const int*)d_in[1];
  const float* kcaches = (const float*)d_in[4];
  const float* vcaches = (const float*)d_in[5];
  const float* emb = (const float*)d_in[6];
  const float* w_qkv = (const float*)d_in[7];
  const float* w_o = (const float*)d_in[8];
  const float* w_gate = (const float*)d_in[9];
  const float* w_up = (const float*)d_in[10];
  const float* w_down = (const float*)d_in[11];
  const float* ln1 = (const float*)d_in[12];
  const float* ln2 = (const float*)d_in[13];
  const float* fln = (const float*)d_in[14];
  const float* rc = (const float*)d_in[15];
  const float* rs = (const float*)d_in[16];

  float* ws = (float*)d_ws;
  float* h = ws;                                  // B*H
  float* xt = h + (size_t)BB * HH;                // H*16
  float* qkv = xt + (size_t)HH * 16;              // B*2560
  float* qt = qkv + (size_t)BB * QKV_N;           // B*KVH*D*16
  float* kn = qt + (size_t)BB * KVH * DD * 16;    // B*KVH*D
  float* vn = kn + (size_t)BB * KVH * DD;         // B*KVH*D
  float* pt = vn + (size_t)BB * KVH * DD;         // B*KVH*SP*16
  float* at = pt + (size_t)BB * KVH * SP * 16;    // H*16
  float* gb = at + (size_t)HH * 16;               // B*I
  float* ub = gb + (size_t)BB * IW;               // B*I
  float* tt = ub + (size_t)BB * IW;               // I*16

  float* logits = (float*)d_out + BB;
  int* ids = (int*)d_out;

  embed_kernel<<<BB, 256, 0, stream>>>(tok, emb, h);
  for (int l = 0; l < NLAYER; ++l) {
    const float* kc_l = kcaches + (size_t)l * BB * SS * KVH * DD;
    const float* vc_l = vcaches + (size_t)l * BB * SS * KVH * DD;
    rmsnorm_kernel<<<BB, 256, 0, stream>>>(h, ln1 + (size_t)l * HH, xt);
    gemm_kernel<HH, 0, false><<<QKV_N / 128, 256, 0, stream>>>(
        xt, w_qkv + (size_t)l * HH * QKV_N, qkv, QKV_N);
    rope_kernel<<<BB, 256, 0, stream>>>(qkv, rc, rs, pos, qt, kn, vn);
    scores_kernel<<<(BB * KVH * (SP / 16)) / 8, 256, 0, stream>>>(qt, kc_l, kn,
                                                                  pt, pos);
    softmax_kernel<<<BB * KVH, 256, 0, stream>>>(pt);
    attn_kernel<<<(BB * KVH * (DD / 16)) / 8, 256, 0, stream>>>(pt, vc_l, vn,
                                                                at, pos);
    gemm_kernel<HH, 2, false><<<HH / 128, 256, 0, stream>>>(
        at, w_o + (size_t)l * HH * HH, h, HH);
    rmsnorm_kernel<<<BB, 256, 0, stream>>>(h, ln2 + (size_t)l * HH, xt);
    gemm_kernel<HH, 1, false><<<IW / 128, 256, 0, stream>>>(
        xt, w_gate + (size_t)l * HH * IW, gb, IW);
    gemm_kernel<HH, 0, false><<<IW / 128, 256, 0, stream>>>(
        xt, w_up + (size_t)l * HH * IW, ub, IW);
    gateup_kernel<<<(BB * IW + 255) / 256, 256, 0, stream>>>(gb, ub, tt);
    gemm_kernel<IW, 2, false><<<HH / 128, 256, 0, stream>>>(
        tt, w_down + (size_t)l * IW * HH, h, HH);
  }
  rmsnorm_kernel<<<BB, 256, 0, stream>>>(h, fln, xt);
  gemm_kernel<HH, 0, true><<<VOCAB / 128, 256, 0, stream>>>(xt, emb, logits,
                                                            VOCAB);
  argmax_kernel<<<BB, 256, 0, stream>>>(logits, ids);
}